// SparseLinear_50525995270225
// MI455X (gfx1250) — compile-verified
//
#include <hip/hip_runtime.h>
#include <hip/hip_bf16.h>

// ---------------------------------------------------------------------------
// SparseLinear on MI455X (gfx1250):
//   densify COO -> f32 dense W in d_ws, then dense f32 WMMA GEMM with
//   double-buffered LDS staged via GLOBAL_LOAD_ASYNC_TO_LDS_B128 (ASYNCcnt).
//   Whole working set (x 16MB + W 64MB + out 16MB) fits in 192MB L2.
// ---------------------------------------------------------------------------

#define IN_F   4096
#define OUT_F  4096
#define BATCH  1024

typedef __attribute__((ext_vector_type(2))) float v2f;
typedef __attribute__((ext_vector_type(8))) float v8f;
typedef __attribute__((ext_vector_type(4))) int   v4i;

// ---------------- kernel 1: zero the dense-W scratch -----------------------
__global__ void zero_f32(float4* __restrict__ p, int n4) {
    int i = blockIdx.x * blockDim.x + threadIdx.x;
    if (i < n4) p[i] = make_float4(0.f, 0.f, 0.f, 0.f);
}

// ---------------- kernel 2: COO scatter-add into dense W -------------------
__global__ void coo_scatter(const float* __restrict__ vals,
                            const int*   __restrict__ rows,
                            const int*   __restrict__ cols,
                            float*       __restrict__ Wd,
                            int nnz) {
    int i = blockIdx.x * blockDim.x + threadIdx.x;
    if (i < nnz) {
        // duplicates must accumulate (reference uses .add)
        atomicAdd(&Wd[(long)rows[i] * IN_F + cols[i]], vals[i]);
    }
}

// ---------------- async global->LDS helpers (gfx1250) ----------------------
// Probe result (round 2 diagnostic): builtin param 0 is `int4 addrspace(1)*`.
typedef __attribute__((address_space(1))) v4i as1_v4i;
typedef __attribute__((address_space(3))) v4i as3_v4i;

__device__ __forceinline__ void async_copy_b128(const float* gsrc, float* ldst) {
#if __has_builtin(__builtin_amdgcn_global_load_async_to_lds_b128)
    __builtin_amdgcn_global_load_async_to_lds_b128(
        (as1_v4i*)gsrc, (as3_v4i*)ldst, /*offset=*/0, /*cpol=*/0);
#else
    // CDNA5 ISA: GLOBAL_LOAD_ASYNC_TO_LDS_B128, GV mode.
    unsigned lds_off = (unsigned)(unsigned long long)ldst; // addr[31:0] = LDS offset
    asm volatile("global_load_async_to_lds_b128 %0, %1, off"
                 :: "v"(lds_off), "v"(gsrc) : "memory");
#endif
}

__device__ __forceinline__ void wait_async_le8() {
#if __has_builtin(__builtin_amdgcn_s_wait_asynccnt)
    __builtin_amdgcn_s_wait_asynccnt(8);
#else
    asm volatile("s_wait_asynccnt 0x8" ::: "memory");
#endif
}
__device__ __forceinline__ void wait_async_le0() {
#if __has_builtin(__builtin_amdgcn_s_wait_asynccnt)
    __builtin_amdgcn_s_wait_asynccnt(0);
#else
    asm volatile("s_wait_asynccnt 0x0" ::: "memory");
#endif
}

// ---------------- kernel 3: dense f32 WMMA GEMM ----------------------------
// out[b, o] = sum_k x[b,k] * Wd[o,k] + bias[o]
//
// Workgroup: 256 threads = 8 waves. Output tile 128(M=batch) x 128(N=out).
// Wave w covers a 32x64 sub-tile: 2x4 accumulators of 16x16 (v8f each).
// K staged through double-buffered LDS in 32-wide slices via async B128
// copies; inner micro-step K=4 per WMMA.
//
// WMMA 16x16x4 f32 per-lane layout (ISA 7.12.2):
//   A (16x4):  lane l, vgpr v -> (M = l%16, K = 2*(l/16) + v)
//   B (4x16):  lane l, vgpr v -> (N = l%16, K = 2*(l/16) + v)   [symmetric]
//   C/D:       lane l, vgpr v -> (M = v + 8*(l/16), N = l%16)
#define TILE_M 128
#define TILE_N 128
#define KB     32
#define KPAD   36   // row stride (floats): 144B -> 16B-aligned B128 stores,
                    // conflict-free ds_load_2addr_b64 fragment reads

__global__ __launch_bounds__(256)
void wmma_gemm_f32(const float* __restrict__ x,
                   const float* __restrict__ Wd,
                   const float* __restrict__ bias,
                   float*       __restrict__ out) {
    __shared__ float As[2][TILE_M * KPAD];
    __shared__ float Bs[2][TILE_N * KPAD];

    const int tid  = threadIdx.x;
    const int wave = tid >> 5;
    const int lane = tid & 31;
    const int l16  = lane & 15;
    const int lhi  = lane >> 4;            // 0 or 1 -> K half
    const int m0   = (wave & 3) * 32;      // wave M offset inside tile
    const int n0   = (wave >> 2) * 64;     // wave N offset inside tile

    const int tileM = blockIdx.y * TILE_M; // batch offset
    const int tileN = blockIdx.x * TILE_N; // out-feature offset

    // this thread's 4 staging slots: f = tid + i*256 -> row f/8, col (f%8)*4
    const int r_[4] = { tid >> 3, (tid + 256) >> 3, (tid + 512) >> 3, (tid + 768) >> 3 };
    const int c_   = (tid & 7) << 2;       // same col group for all 4 (tid%8 invariant)

    v8f acc[2][4];
#pragma unroll
    for (int mi = 0; mi < 2; ++mi)
#pragma unroll
        for (int ni = 0; ni < 4; ++ni)
            acc[mi][ni] = (v8f){0.f, 0.f, 0.f, 0.f, 0.f, 0.f, 0.f, 0.f};

    // ---- prologue: stage k0 = 0 into buffer 0 (8 async B128 per thread) ----
#pragma unroll
    for (int i = 0; i < 4; ++i) {
        async_copy_b128(&x [(long)(tileM + r_[i]) * IN_F + c_], &As[0][r_[i] * KPAD + c_]);
        async_copy_b128(&Wd[(long)(tileN + r_[i]) * IN_F + c_], &Bs[0][r_[i] * KPAD + c_]);
    }

    int s = 0;
    for (int k0 = 0; k0 < IN_F; k0 += KB) {
        // ---- prefetch next K-slice into the other buffer, then wait for ours
        if (k0 + KB < IN_F) {
            const int kn = k0 + KB;
#pragma unroll
            for (int i = 0; i < 4; ++i) {
                async_copy_b128(&x [(long)(tileM + r_[i]) * IN_F + kn + c_],
                                &As[s ^ 1][r_[i] * KPAD + c_]);
                async_copy_b128(&Wd[(long)(tileN + r_[i]) * IN_F + kn + c_],
                                &Bs[s ^ 1][r_[i] * KPAD + c_]);
            }
            wait_async_le8();  // in-order completion: our buffer-s fills are done
        } else {
            wait_async_le0();
        }
        __syncthreads();       // buffer s visible to all waves

        const float* Asb = As[s];
        const float* Bsb = Bs[s];
#pragma unroll
        for (int kk = 0; kk < KB; kk += 4) {
            const int ka = kk + 2 * lhi;   // this lane's K pair
            v2f a[2], b[4];
#pragma unroll
            for (int mi = 0; mi < 2; ++mi)
                a[mi] = *reinterpret_cast<const v2f*>(
                    &Asb[(m0 + mi * 16 + l16) * KPAD + ka]);
#pragma unroll
            for (int ni = 0; ni < 4; ++ni)
                b[ni] = *reinterpret_cast<const v2f*>(
                    &Bsb[(n0 + ni * 16 + l16) * KPAD + ka]);
#pragma unroll
            for (int mi = 0; mi < 2; ++mi)
#pragma unroll
                for (int ni = 0; ni < 4; ++ni)
                    acc[mi][ni] = __builtin_amdgcn_wmma_f32_16x16x4_f32(
                        /*neg_a=*/false, a[mi],
                        /*neg_b=*/false, b[ni],
                        /*c_mod=*/(short)0, acc[mi][ni],
                        /*reuse_a=*/false, /*reuse_b=*/false);
        }
        __syncthreads();       // all waves done reading buffer s -> reusable
        s ^= 1;
    }

    // ---- epilogue: add bias, write out ----
#pragma unroll
    for (int mi = 0; mi < 2; ++mi) {
#pragma unroll
        for (int ni = 0; ni < 4; ++ni) {
            const int nCol = tileN + n0 + ni * 16 + l16;
            const float bv = bias[nCol];
#pragma unroll
            for (int v = 0; v < 8; ++v) {
                const int mRow = tileM + m0 + mi * 16 + v + 8 * lhi;
                out[(long)mRow * OUT_F + nCol] = acc[mi][ni][v] + bv;
            }
        }
    }
}

// ---------------------------------------------------------------------------
extern "C" void kernel_launch(void* const* d_in, const int* in_sizes, int n_in,
                              void* d_out, int out_size, void* d_ws, size_t ws_size,
                              hipStream_t stream) {
    const float* x      = (const float*)d_in[0];   // [1024, 4096]
    const float* w_vals = (const float*)d_in[1];   // [NNZ]
    const float* bias   = (const float*)d_in[2];   // [4096]
    const int*   w_rows = (const int*)  d_in[3];   // [NNZ]
    const int*   w_cols = (const int*)  d_in[4];   // [NNZ]
    float*       out    = (float*)d_out;           // [1024, 4096]
    float*       Wd     = (float*)d_ws;            // dense W: 4096*4096 f32 = 64 MB

    const int nnz = in_sizes[1];

    // 1) zero dense-W scratch (d_ws is poisoned; must re-zero every call)
    {
        const int n4 = (OUT_F * IN_F) / 4;          // 4,194,304 float4
        dim3 grid((n4 + 255) / 256);
        zero_f32<<<grid, 256, 0, stream>>>((float4*)Wd, n4);
    }
    // 2) scatter COO into dense W (L2-resident atomics)
    {
        dim3 grid((nnz + 255) / 256);
        coo_scatter<<<grid, 256, 0, stream>>>(w_vals, w_rows, w_cols, Wd, nnz);
    }
    // 3) dense f32 WMMA GEMM + bias
    {
        dim3 grid(OUT_F / TILE_N, BATCH / TILE_M);  // (32, 8)
        wmma_gemm_f32<<<grid, 256, 0, stream>>>(x, Wd, bias, out);
    }
}